// YOLOv5_20796231647911
// MI455X (gfx1250) — compile-verified
//
#include <hip/hip_runtime.h>
#include <stdint.h>

#define USE_ASYNC_LDS 1

namespace {
constexpr int B = 16, N = 25200, NC = 80, ROW = NC + 5;   // pred [B,N,85]
constexpr int K = 4096, MAXDET = 300, BCAP = 2048;        // boundary-bin capacity (expected ~4 used)
constexpr float CONF = 0.25f, IOU_T = 0.45f, MAX_WH = 7680.0f, NEGV = -1000000000.0f;
constexpr int HBINS = 32768;
constexpr unsigned HI_MIN = 0x3E800000u >> 9;   // bits(0.25)>>9 = 0x1F4000
// k_nms dynamic LDS layout
constexpr size_t SC_OFF  = 0;                       // K floats        (16384 B)
constexpr size_t BX_OFF  = SC_OFF + (size_t)K * 4;  // K float4        (65536 B)
constexpr size_t RED_OFF = BX_OFF + (size_t)K * 16; // 32 u64          (256 B)
constexpr size_t PI_OFF  = RED_OFF + 256;           // MAXDET u32      (1200 B)
constexpr size_t PV_OFF  = PI_OFF + 1216;           // MAXDET u32      (1200 B)
constexpr size_t NMS_SMEM = PV_OFF + 1216;
}

// ---------- CDNA5 async global->LDS helpers ----------
__device__ __forceinline__ void async_g2l_b128(void* ldsPtr, const void* gPtr) {
#if USE_ASYNC_LDS
  unsigned loff = (unsigned)(uintptr_t)ldsPtr;   // generic LDS addr: low 32 bits = LDS offset
  asm volatile("global_load_async_to_lds_b128 %0, %1, off"
               :: "v"(loff), "v"(gPtr) : "memory");
#else
  *(float4*)ldsPtr = *(const float4*)gPtr;
#endif
}
__device__ __forceinline__ void wait_async_lds() {
#if USE_ASYNC_LDS
#if __has_builtin(__builtin_amdgcn_s_wait_asynccnt)
  __builtin_amdgcn_s_wait_asynccnt(0);
#else
  asm volatile("s_wait_asynccnt 0" ::: "memory");
#endif
#endif
}

__device__ __forceinline__ unsigned long long shfl_xor_u64(unsigned long long v, int m) {
  unsigned lo = (unsigned)v, hi = (unsigned)(v >> 32);
  lo = __shfl_xor(lo, m, 32);
  hi = __shfl_xor(hi, m, 32);
  return ((unsigned long long)hi << 32) | lo;
}

template <int NN>
__device__ __forceinline__ void bitonic_desc_u64(unsigned long long* a) {
  for (int k2 = 2; k2 <= NN; k2 <<= 1)
    for (int j = k2 >> 1; j > 0; j >>= 1) {
      __syncthreads();
      for (int i = threadIdx.x; i < NN; i += blockDim.x) {
        int x = i ^ j;
        if (x > i) {
          unsigned long long A = a[i], C = a[x];
          bool up = ((i & k2) == 0);
          if ((A < C) == up) { a[i] = C; a[x] = A; }  // descending
        }
      }
    }
  __syncthreads();
}

// ---------- Pass 1: coarse 15-bit histogram of passing score bits ----------
__global__ void k_hist1(const float* __restrict__ pred, unsigned* __restrict__ hist) {
  int g = blockIdx.x * blockDim.x + threadIdx.x;
  if (g >= B * N) return;
  int b = g / N;
  const float* p = pred + (size_t)g * ROW;
  float obj = p[4];
  unsigned* h = hist + (size_t)b * HBINS;
  for (int c = 0; c < NC; ++c) {
    float s = obj * p[5 + c];
    if (s > CONF) {
      unsigned bits = __float_as_uint(s);
      atomicAdd(&h[(bits >> 9) - HI_MIN], 1u);
    }
  }
}

// ---------- Find coarse bin holding the K-th largest ----------
__global__ void k_findHi(const unsigned* __restrict__ hist, unsigned* __restrict__ sel) {
  int b = blockIdx.x;
  if (threadIdx.x != 0) return;
  const unsigned* h = hist + (size_t)b * HBINS;
  unsigned cum = 0; int hiBin = -1; unsigned above = 0;
  for (int i = HBINS - 1; i >= 0; --i) {
    unsigned c = h[i];
    if (cum + c >= (unsigned)K) { hiBin = i; above = cum; break; }
    cum += c;
  }
  sel[b * 2 + 0] = (unsigned)hiBin;                 // 0xFFFFFFFF => fewer than K pass
  sel[b * 2 + 1] = (hiBin < 0) ? cum : above;
}

// ---------- Pass 2: compact definite top (hi > hiBin) + boundary bin (hi == hiBin) ----------
__global__ void k_compact(const float* __restrict__ pred, const unsigned* __restrict__ sel,
                          unsigned* __restrict__ cnt,
                          unsigned long long* __restrict__ gtKeys,
                          unsigned long long* __restrict__ bndKeys) {
  int g = blockIdx.x * blockDim.x + threadIdx.x;
  if (g >= B * N) return;
  int b = g / N, n = g % N;
  unsigned hiBin = sel[b * 2 + 0];
  const float* p = pred + (size_t)g * ROW;
  float obj = p[4];
  for (int c = 0; c < NC; ++c) {
    float s = obj * p[5 + c];
    if (s > CONF) {
      unsigned bits = __float_as_uint(s);
      unsigned hi = (bits >> 9) - HI_MIN;
      unsigned flat = (unsigned)(n * NC + c);
      unsigned long long key = ((unsigned long long)bits << 32) | (unsigned)(~flat);
      if (hiBin == 0xFFFFFFFFu || hi > hiBin) {
        unsigned pos = atomicAdd(&cnt[b * 2 + 0], 1u);
        if (pos < (unsigned)K) gtKeys[(size_t)b * K + pos] = key;
      } else if (hi == hiBin) {
        unsigned pos = atomicAdd(&cnt[b * 2 + 1], 1u);
        if (pos < (unsigned)BCAP) bndKeys[(size_t)b * BCAP + pos] = key;
      }
    }
  }
}

// ---------- Build top-K list in exact jax.lax.top_k order; emit boxes ----------
__global__ __launch_bounds__(1024) void k_finalize(
    const float* __restrict__ pred, const unsigned* __restrict__ sel,
    const unsigned* __restrict__ cnt,
    const unsigned long long* __restrict__ gtKeys,
    const unsigned long long* __restrict__ bndKeys,
    float* __restrict__ candScore, float* __restrict__ candCls,
    float4* __restrict__ cboxes, float4* __restrict__ oboxes) {
  __shared__ unsigned long long kk[K];     // 32 KB
  __shared__ unsigned long long bb[BCAP];  // 16 KB
  int b = blockIdx.x, tid = threadIdx.x;
  unsigned hiBin = sel[b * 2 + 0];
  unsigned gtc = cnt[b * 2 + 0]; if (gtc > (unsigned)K) gtc = K;
  unsigned need = (hiBin == 0xFFFFFFFFu) ? 0u : (unsigned)K - gtc;

  // Async-stage both key buffers (zero-filled beyond live counts by host memset)
  {
    const unsigned long long* ggt = gtKeys + (size_t)b * K;
    const unsigned long long* gbd = bndKeys + (size_t)b * BCAP;
    for (int r = 0; r < 2; ++r) {            // 4096 u64 = 2048 b128
      int i = tid + r * 1024;
      async_g2l_b128(&kk[i * 2], &ggt[i * 2]);
    }
    async_g2l_b128(&bb[tid * 2], &gbd[tid * 2]);  // 2048 u64 = 1024 b128
    wait_async_lds();
  }
  __syncthreads();

  bitonic_desc_u64<K>(kk);      // score desc, original index asc (via ~idx)
  bitonic_desc_u64<BCAP>(bb);

  for (int s = tid; s < K; s += 1024) {
    #pragma clang fp contract(off)
    unsigned long long key;
    if (s < (int)gtc)                 key = kk[s];
    else if (s < (int)(gtc + need))   key = ((s - (int)gtc) < BCAP) ? bb[s - gtc] : 0ull;
    else                              key = 0ull;
    float score; unsigned flat;
    if (key != 0ull) { score = __uint_as_float((unsigned)(key >> 32)); flat = ~(unsigned)key; }
    else             { score = NEGV; flat = 0u; }   // filler: never valid, output zeroed
    unsigned anchor = flat / NC, cls = flat % NC;
    const float* p = pred + ((size_t)b * N + anchor) * ROW;
    float x = p[0], y = p[1], w = p[2], h = p[3];
    float hw = w * 0.5f, hh = h * 0.5f;
    float4 cb = make_float4(x - hw, y - hh, x + hw, y + hh);
    float off = (float)cls * MAX_WH;
    float4 ob = make_float4(cb.x + off, cb.y + off, cb.z + off, cb.w + off);
    size_t o = (size_t)b * K + s;
    candScore[o] = score; candCls[o] = (float)cls; cboxes[o] = cb; oboxes[o] = ob;
  }
}

// ---------- Greedy NMS (LDS-resident) + fused output gather ----------
__global__ __launch_bounds__(1024) void k_nms(
    const float* __restrict__ candScore, const float4* __restrict__ oboxes,
    const float4* __restrict__ cboxes, const float* __restrict__ candCls,
    float* __restrict__ out) {
  extern __shared__ unsigned char smemRaw[];
  float* sc = (float*)(smemRaw + SC_OFF);                       // K scores
  float4* bx = (float4*)(smemRaw + BX_OFF);                     // K offset boxes
  unsigned long long* red = (unsigned long long*)(smemRaw + RED_OFF);  // 32 wave partials
  unsigned* pickI = (unsigned*)(smemRaw + PI_OFF);
  unsigned* pickV = (unsigned*)(smemRaw + PV_OFF);
  int b = blockIdx.x, tid = threadIdx.x;
  int lane = tid & 31, wave = tid >> 5;
  const float* gsc = candScore + (size_t)b * K;
  const float4* gbx = oboxes + (size_t)b * K;

  // Stage candidates into LDS with gfx1250 async copies (ASYNCcnt-tracked)
  {
    int i0 = tid * 4;                       // 1024 threads * 4 floats = K scores
    async_g2l_b128(&sc[i0], &gsc[i0]);
    for (int r = 0; r < 4; ++r) {           // 4096 float4 boxes
      int i = tid + r * 1024;
      async_g2l_b128(&bx[i], &gbx[i]);
    }
    wait_async_lds();
  }
  __syncthreads();

  for (int j = 0; j < MAXDET; ++j) {
    // block argmax, exact first-occurrence tie-break: key = (ordered_bits, ~slot)
    unsigned long long best = 0ull;
    for (int i = tid; i < K; i += 1024) {
      unsigned bbits = __float_as_uint(sc[i]);
      unsigned ord = bbits ^ ((bbits & 0x80000000u) ? 0xFFFFFFFFu : 0x80000000u);
      unsigned long long key = ((unsigned long long)ord << 32) | (unsigned)(~(unsigned)i);
      if (key > best) best = key;
    }
    // wave32 shuffle reduce (5 steps), then 32 partials, then wave-0 reduce
    for (int m = 16; m > 0; m >>= 1) {
      unsigned long long o = shfl_xor_u64(best, m);
      if (o > best) best = o;
    }
    if (lane == 0) red[wave] = best;
    __syncthreads();
    if (wave == 0) {
      unsigned long long v = red[lane];
      for (int m = 16; m > 0; m >>= 1) {
        unsigned long long o = shfl_xor_u64(v, m);
        if (o > v) v = o;
      }
      if (lane == 0) red[0] = v;
    }
    __syncthreads();
    unsigned wi = ~(unsigned)red[0];
    float swin = sc[wi];
    bool sv = swin > CONF;
    float4 wb = bx[wi];
    __syncthreads();   // everyone has read sc[wi]/bx before suppression writes

    if (sv) {
      float areaB = (wb.z - wb.x) * (wb.w - wb.y);
      for (int i = tid; i < K; i += 1024) {
        #pragma clang fp contract(off)
        float4 c = bx[i];
        float ltx = fmaxf(wb.x, c.x), lty = fmaxf(wb.y, c.y);
        float rbx = fminf(wb.z, c.z), rby = fminf(wb.w, c.w);
        float iw = rbx - ltx; iw = iw > 0.0f ? iw : 0.0f;
        float ih = rby - lty; ih = ih > 0.0f ? ih : 0.0f;
        float inter = iw * ih;
        float areaC = (c.z - c.x) * (c.w - c.y);
        float iou = inter / (areaB + areaC - inter + 1e-7f);
        if (iou > IOU_T) sc[i] = NEGV;
      }
    }
    if (tid == 0) {
      sc[wi] = NEGV;   // picked slot always removed, even when invalid
      pickI[j] = wi;
      pickV[j] = sv ? 1u : 0u;
    }
    __syncthreads();
  }

  // fused gather: [box, score, cls] * valid
  for (int j = tid; j < MAXDET; j += 1024) {
    unsigned wi = pickI[j];
    float v = pickV[j] ? 1.0f : 0.0f;
    size_t o = (size_t)b * K + wi;
    float4 cb = cboxes[o];
    float* dst = out + ((size_t)b * MAXDET + j) * 6;
    dst[0] = cb.x * v; dst[1] = cb.y * v; dst[2] = cb.z * v; dst[3] = cb.w * v;
    dst[4] = candScore[o] * v;
    dst[5] = candCls[o] * v;
  }
}

extern "C" void kernel_launch(void* const* d_in, const int* /*in_sizes*/, int /*n_in*/,
                              void* d_out, int /*out_size*/, void* d_ws, size_t /*ws_size*/,
                              hipStream_t stream) {
  const float* pred = (const float*)d_in[0];
  float* out = (float*)d_out;

  char* ws = (char*)d_ws;
  size_t off = 0;
  auto take = [&](size_t bytes) -> char* {
    char* p = ws + off;
    off += (bytes + 255) & ~(size_t)255;
    return p;
  };
  unsigned* hist1 = (unsigned*)take((size_t)B * HBINS * 4);
  unsigned* cnt   = (unsigned*)take((size_t)B * 2 * 4);
  unsigned* sel   = (unsigned*)take((size_t)B * 2 * 4);
  unsigned long long* gtKeys  = (unsigned long long*)take((size_t)B * K * 8);
  unsigned long long* bndKeys = (unsigned long long*)take((size_t)B * BCAP * 8);
  float* candScore = (float*)take((size_t)B * K * 4);
  float* candCls   = (float*)take((size_t)B * K * 4);
  float4* cboxes   = (float4*)take((size_t)B * K * 16);
  float4* oboxes   = (float4*)take((size_t)B * K * 16);

  hipMemsetAsync(hist1,   0, (size_t)B * HBINS * 4, stream);
  hipMemsetAsync(cnt,     0, (size_t)B * 2 * 4, stream);
  hipMemsetAsync(gtKeys,  0, (size_t)B * K * 8, stream);     // key 0 == filler
  hipMemsetAsync(bndKeys, 0, (size_t)B * BCAP * 8, stream);

  const int threads = 256;
  const int blocks = (B * N + threads - 1) / threads;
  k_hist1<<<blocks, threads, 0, stream>>>(pred, hist1);
  k_findHi<<<B, 32, 0, stream>>>(hist1, sel);
  k_compact<<<blocks, threads, 0, stream>>>(pred, sel, cnt, gtKeys, bndKeys);
  k_finalize<<<B, 1024, 0, stream>>>(pred, sel, cnt, gtKeys, bndKeys,
                                     candScore, candCls, cboxes, oboxes);

  hipFuncSetAttribute((const void*)k_nms, hipFuncAttributeMaxDynamicSharedMemorySize,
                      (int)NMS_SMEM);
  k_nms<<<B, 1024, NMS_SMEM, stream>>>(candScore, oboxes, cboxes, candCls, out);
}